// KernelAttention_8297876815994
// MI455X (gfx1250) — compile-verified
//
#include <hip/hip_runtime.h>

// Problem constants (match reference)
#define BS     2
#define SEQ    1024
#define DMODEL 512
#define NHEADS 8
#define HD     64
#define NCLS   1000
#define BH     (BS * NHEADS)
#define MROWS  (BS * SEQ)   // 2048

typedef __attribute__((ext_vector_type(2))) float v2f;
typedef __attribute__((ext_vector_type(8))) float v8f;

// One fp32 WMMA step: D(16x16) += A(16x4) * B(4x16), fp32 throughout.
__device__ __forceinline__ v8f wmma_f32_step(v2f a, v2f b, v8f c) {
    // 8 args: (neg_a, A, neg_b, B, c_mod, C, reuse_a, reuse_b)
    return __builtin_amdgcn_wmma_f32_16x16x4_f32(
        false, a, false, b, (short)0, c, false, false);
}

// Feed an 8-wide K chunk (one float4 per lane for A and B) into two WMMA steps.
// K->hardware-slot mapping is a fixed permutation applied identically to A and B,
// so the reduction is exact: half-0 lanes carry mem-K [kk,kk+4), half-1 [kk+4,kk+8).
__device__ __forceinline__ v8f wmma_chunk8(float4 a, float4 b, v8f c) {
    v2f a0; a0.x = a.x; a0.y = a.y;
    v2f b0; b0.x = b.x; b0.y = b.y;
    c = wmma_f32_step(a0, b0, c);
    v2f a1; a1.x = a.z; a1.y = a.w;
    v2f b1; b1.x = b.z; b1.y = b.w;
    return wmma_f32_step(a1, b1, c);
}

// ---------------------------------------------------------------------------
// GEMM: Y[M x N] = X[M x K] @ W[N x K]^T + bias[N]
// One wave computes a 16(M) x 64(N) tile: 4 accumulators, A reused 4x.
// Per 8-K chunk: 1 b128 A-load + 4 b128 B-loads + 8 WMMAs.
// N edge (N=1000): OOB W rows clamped to row N-1 (garbage columns computed but
// never stored) -- no EXEC manipulation anywhere near the WMMAs.
// ---------------------------------------------------------------------------
__global__ void gemm_xwt_kernel(const float* __restrict__ X,
                                const float* __restrict__ W,
                                const float* __restrict__ bias,
                                float* __restrict__ Y,
                                int M, int N, int K,
                                int ntilesN64, int totalTiles) {
    int tile = blockIdx.x * blockDim.y + threadIdx.y;   // wave-uniform
    if (tile >= totalTiles) return;
    int tm = tile / ntilesN64;
    int tn = tile - tm * ntilesN64;
    int m0 = tm << 4, n0 = tn << 6;

    int l    = threadIdx.x;
    int half = l >> 4;         // 0: lanes 0-15, 1: lanes 16-31
    int lm   = l & 15;

    const float* xr = X + (size_t)(m0 + lm) * K + 4 * half;   // A: row m0+lm

    const float* wr[4];
    int ncol[4];
    #pragma unroll
    for (int j = 0; j < 4; ++j) {
        int n = n0 + 16 * j + lm;
        ncol[j] = n;
        int nc = n < N ? n : (N - 1);                 // clamp OOB rows
        wr[j] = W + (size_t)nc * K + 4 * half;
    }

    v8f acc[4] = {{}, {}, {}, {}};
    for (int kk = 0; kk < K; kk += 8) {
        float4 av = *reinterpret_cast<const float4*>(xr + kk);
        #pragma unroll
        for (int j = 0; j < 4; ++j) {
            float4 bv = *reinterpret_cast<const float4*>(wr[j] + kk);
            acc[j] = wmma_chunk8(av, bv, acc[j]);
        }
    }

    #pragma unroll
    for (int j = 0; j < 4; ++j) {
        int n = ncol[j];
        if (n < N) {
            float bvv = bias ? bias[n] : 0.0f;
            #pragma unroll
            for (int r = 0; r < 8; ++r) {
                int m = m0 + r + 8 * half;   // D layout: reg r -> M=r (+8 upper half)
                Y[(size_t)m * N + n] = acc[j][r] + bvv;
            }
        }
    }
}

// ---------------------------------------------------------------------------
// Per-head row norms: out[bh*SEQ + s] = sum_d P[b, s, h*HD + d]^2
// ---------------------------------------------------------------------------
__global__ void rownorm_kernel(const float* __restrict__ P, float* __restrict__ out) {
    int idx = blockIdx.x * blockDim.x + threadIdx.x;
    if (idx >= BH * SEQ) return;
    int bh = idx / SEQ, s = idx - bh * SEQ;
    int b = bh / NHEADS, h = bh - b * NHEADS;
    const float* row = P + ((size_t)(b * SEQ + s)) * DMODEL + h * HD;
    float acc = 0.0f;
    #pragma unroll 8
    for (int d = 0; d < HD; ++d) { float v = row[d]; acc += v * v; }
    out[idx] = acc;
}

// ---------------------------------------------------------------------------
// M[bh, s, t] = exp(-0.5 * max(qsq+ksq-2*q.k, 0) / ls^2) - lambda*(s==t)
// One wave per 16(s) x 64(t) tile; cross term over HD=64 = 8 chunks x 8 WMMAs.
// ---------------------------------------------------------------------------
__global__ void rbf_m_kernel(const float* __restrict__ Q,
                             const float* __restrict__ Kp,
                             const float* __restrict__ qsq,
                             const float* __restrict__ ksq,
                             const float* __restrict__ ls_p,
                             const float* __restrict__ lam_p,
                             float* __restrict__ Mmat,
                             int totalTiles) {
    int tile = blockIdx.x * blockDim.y + threadIdx.y;
    if (tile >= totalTiles) return;
    const int ntT = SEQ / 64;            // 16 t-groups
    const int ntS = SEQ / 16;            // 64 s-tiles
    int tt  = tile % ntT;
    int tmp = tile / ntT;
    int ts  = tmp % ntS;
    int bh  = tmp / ntS;
    int b = bh / NHEADS, h = bh - b * NHEADS;
    int s0 = ts << 4, t0 = tt << 6;

    int l = threadIdx.x, half = l >> 4, lm = l & 15;

    const float* qr = Q + ((size_t)(b * SEQ + s0 + lm)) * DMODEL + h * HD + 4 * half;
    const float* kr[4];
    #pragma unroll
    for (int j = 0; j < 4; ++j)
        kr[j] = Kp + ((size_t)(b * SEQ + t0 + 16 * j + lm)) * DMODEL + h * HD + 4 * half;

    v8f acc[4] = {{}, {}, {}, {}};
    #pragma unroll
    for (int kk = 0; kk < HD; kk += 8) {
        float4 av = *reinterpret_cast<const float4*>(qr + kk);
        #pragma unroll
        for (int j = 0; j < 4; ++j) {
            float4 bv = *reinterpret_cast<const float4*>(kr[j] + kk);
            acc[j] = wmma_chunk8(av, bv, acc[j]);
        }
    }

    float lsv = ls_p[0];
    float sc  = 0.5f / (lsv * lsv);
    float lam = lam_p[0];

    float qs[8];
    #pragma unroll
    for (int r = 0; r < 8; ++r)
        qs[r] = qsq[(size_t)bh * SEQ + (s0 + r + 8 * half)];

    float* Mb = Mmat + (size_t)bh * SEQ * SEQ;
    #pragma unroll
    for (int j = 0; j < 4; ++j) {
        int   t   = t0 + 16 * j + lm;
        float kst = ksq[(size_t)bh * SEQ + t];
        #pragma unroll
        for (int r = 0; r < 8; ++r) {
            int   s  = s0 + r + 8 * half;
            float sq = qs[r] + kst - 2.0f * acc[j][r];
            sq = sq < 0.0f ? 0.0f : sq;
            float val = expf(-sq * sc);
            if (s == t) val -= lam;                  // M = K - lambda*I
            Mb[(size_t)s * SEQ + t] = val;
        }
    }
}

// ---------------------------------------------------------------------------
// Y[bh, s, d] = V[b, s, h*HD + d]   (initial RHS for the solve)
// ---------------------------------------------------------------------------
__global__ void vhead_copy_kernel(const float* __restrict__ V, float* __restrict__ Yg) {
    int idx = blockIdx.x * blockDim.x + threadIdx.x;
    if (idx >= BH * SEQ * HD) return;
    int d  = idx % HD;
    int t  = idx / HD;
    int s  = t % SEQ;
    int bh = t / SEQ;
    int b = bh / NHEADS, h = bh - b * NHEADS;
    Yg[idx] = V[((size_t)(b * SEQ + s)) * DMODEL + h * HD + d];
}

// ---------------------------------------------------------------------------
// Batched Gauss-Jordan solve:  M * Y = V  (M destroyed, Y <- M^{-1} V).
// One 1024-thread workgroup per (b,h). Thread t owns one augmented column.
// Pivot row and multiplier column staged in LDS; global accesses coalesced.
// attn = I + lambda * M^{-1}, so only Y is needed downstream.
// ---------------------------------------------------------------------------
__global__ void __launch_bounds__(1024) gj_solve_kernel(float* __restrict__ Mg,
                                                        float* __restrict__ Yg) {
    int bh = blockIdx.x;
    float* Mm = Mg + (size_t)bh * SEQ * SEQ;
    float* Yy = Yg + (size_t)bh * SEQ * HD;

    __shared__ float prow[SEQ + HD];   // normalized pivot row (cols > k) + Y row
    __shared__ float fvec[SEQ];        // multipliers M[i][k]

    int tid = threadIdx.x;

    for (int k = 0; k < SEQ; ++k) {
        int ncolsM = SEQ - k - 1;
        int ncols  = ncolsM + HD;

        // Stage multipliers (column k) and normalized pivot row into LDS.
        fvec[tid] = Mm[(size_t)tid * SEQ + k];
        float invp = 1.0f / Mm[(size_t)k * SEQ + k];
        if (tid < ncols) {
            if (tid < ncolsM) {
                prow[tid] = Mm[(size_t)k * SEQ + (k + 1 + tid)] * invp;
            } else {
                int d = tid - ncolsM;
                float v = Yy[(size_t)k * HD + d] * invp;
                prow[tid] = v;
                Yy[(size_t)k * HD + d] = v;       // row k of Y scaled by 1/pivot
            }
        }
        __syncthreads();

        // Eliminate column k from every other row (above and below: Gauss-Jordan).
        if (tid < ncols) {
            float pv = prow[tid];
            if (tid < ncolsM) {
                int j = k + 1 + tid;
                for (int i = 0; i < SEQ; ++i) {
                    if (i == k) continue;
                    Mm[(size_t)i * SEQ + j] -= fvec[i] * pv;
                }
            } else {
                int d = tid - ncolsM;
                for (int i = 0; i < SEQ; ++i) {
                    if (i == k) continue;
                    Yy[(size_t)i * HD + d] -= fvec[i] * pv;
                }
            }
        }
        __syncthreads();
    }
}

// ---------------------------------------------------------------------------
// ctx[b, s, h*HD+d] = V[b, s, h*HD+d] + lambda * Y[bh, s, d]
// (attn @ v = v + lambda * M^{-1} v)
// ---------------------------------------------------------------------------
__global__ void combine_kernel(const float* __restrict__ V,
                               const float* __restrict__ Yg,
                               const float* __restrict__ lam_p,
                               float* __restrict__ ctx) {
    int idx = blockIdx.x * blockDim.x + threadIdx.x;
    if (idx >= BS * SEQ * DMODEL) return;
    int e = idx % DMODEL;
    int t = idx / DMODEL;
    int s = t % SEQ;
    int b = t / SEQ;
    int h = e / HD, d = e - h * HD;
    float lam = lam_p[0];
    ctx[idx] = V[idx] + lam * Yg[(((size_t)(b * NHEADS + h)) * SEQ + s) * HD + d];
}

// ---------------------------------------------------------------------------
// Host-side orchestration
// ---------------------------------------------------------------------------
extern "C" void kernel_launch(void* const* d_in, const int* in_sizes, int n_in,
                              void* d_out, int out_size, void* d_ws, size_t ws_size,
                              hipStream_t stream) {
    const float* x   = (const float*)d_in[0];
    const float* Wq  = (const float*)d_in[1];
    const float* bq  = (const float*)d_in[2];
    const float* Wk  = (const float*)d_in[3];
    const float* bk  = (const float*)d_in[4];
    const float* Wv  = (const float*)d_in[5];
    const float* bv  = (const float*)d_in[6];
    const float* Wo  = (const float*)d_in[7];
    const float* bo  = (const float*)d_in[8];
    const float* ls  = (const float*)d_in[9];
    const float* lam = (const float*)d_in[10];
    float* out = (float*)d_out;

    // Workspace carve-up (floats): Q, K, V, qsq, ksq, Y, ctx, M  (~88 MB total)
    float* Q    = (float*)d_ws;
    float* Kp   = Q    + (size_t)MROWS * DMODEL;
    float* Vp   = Kp   + (size_t)MROWS * DMODEL;
    float* qsq  = Vp   + (size_t)MROWS * DMODEL;
    float* ksq  = qsq  + (size_t)BH * SEQ;
    float* Yg   = ksq  + (size_t)BH * SEQ;
    float* ctx  = Yg   + (size_t)BH * SEQ * HD;
    float* Mmat = ctx  + (size_t)MROWS * DMODEL;

    dim3 gblock(32, 4, 1);   // 4 waves per block, one 16x64 tile per wave

    // 1) QKV projections: [2048x512] @ [512x512]^T + bias
    {
        int ntN64 = DMODEL / 64;                       // 8
        int tiles = (MROWS / 16) * ntN64;              // 1024
        int nblk = (tiles + 3) / 4;
        gemm_xwt_kernel<<<nblk, gblock, 0, stream>>>(x, Wq, bq, Q,  MROWS, DMODEL, DMODEL, ntN64, tiles);
        gemm_xwt_kernel<<<nblk, gblock, 0, stream>>>(x, Wk, bk, Kp, MROWS, DMODEL, DMODEL, ntN64, tiles);
        gemm_xwt_kernel<<<nblk, gblock, 0, stream>>>(x, Wv, bv, Vp, MROWS, DMODEL, DMODEL, ntN64, tiles);
    }

    // 2) Per-head squared row norms of q and k
    {
        int n = BH * SEQ;
        rownorm_kernel<<<(n + 255) / 256, 256, 0, stream>>>(Q,  qsq);
        rownorm_kernel<<<(n + 255) / 256, 256, 0, stream>>>(Kp, ksq);
    }

    // 3) M = exp(-0.5*||q-k||^2 / ls^2) - lambda*I, tiled WMMA cross-term
    {
        int tiles = BH * (SEQ / 16) * (SEQ / 64);      // 16*64*16 = 16384
        rbf_m_kernel<<<(tiles + 3) / 4, gblock, 0, stream>>>(Q, Kp, qsq, ksq, ls, lam, Mmat, tiles);
    }

    // 4) Y <- v_head (RHS), then solve M Y = V per (b,h)
    {
        int n = BH * SEQ * HD;
        vhead_copy_kernel<<<(n + 255) / 256, 256, 0, stream>>>(Vp, Yg);
        gj_solve_kernel<<<BH, 1024, 0, stream>>>(Mmat, Yg);
    }

    // 5) ctx = v + lambda * Y   (== attn @ v)
    {
        int n = BS * SEQ * DMODEL;
        combine_kernel<<<(n + 255) / 256, 256, 0, stream>>>(Vp, Yg, lam, ctx);
    }

    // 6) Final classifier: out[2048x1000] = ctx @ Wo^T + bo (N edge clamped/guarded)
    {
        int ntN64 = (NCLS + 63) / 64;                  // 16 groups (last partial)
        int tiles = (MROWS / 16) * ntN64;              // 2048
        gemm_xwt_kernel<<<(tiles + 3) / 4, gblock, 0, stream>>>(ctx, Wo, bo, out, MROWS, NCLS, DMODEL, ntN64, tiles);
    }
}